// GCN_38182259261567
// MI455X (gfx1250) — compile-verified
//
#include <hip/hip_runtime.h>
#include <math.h>

typedef __attribute__((ext_vector_type(2))) float v2f;
typedef __attribute__((ext_vector_type(8))) float v8f;

#define NF 128          // feature width (all layers)
#define EPS 1e-12f

// ------------------------------------------------------------------
// utility kernels
// ------------------------------------------------------------------
__global__ void k_fill_f32(float* __restrict__ p, float v, int n) {
  int i = blockIdx.x * blockDim.x + threadIdx.x;
  if (i < n) p[i] = v;
}

__global__ void k_deg_count(const int* __restrict__ dst, float* __restrict__ deg, int E) {
  int e = blockIdx.x * blockDim.x + threadIdx.x;
  if (e < E) atomicAdd(&deg[dst[e]], 1.0f);
}

__global__ void k_rsqrt_inplace(float* __restrict__ p, int n) {
  int i = blockIdx.x * blockDim.x + threadIdx.x;
  if (i < n) p[i] = rsqrtf(p[i]);   // deg >= 1 always (self loop), no zero guard needed
}

// ------------------------------------------------------------------
// GEMM: H[N,128] = X[N,128] @ W[128,128] using V_WMMA_F32_16X16X4_F32
// block = 256 threads (8 wave32), each wave owns one 16x16 column tile.
// The 16x128 A tile is staged into (padded) LDS via CDNA5 async
// memory->LDS transfers (GLOBAL_LOAD_ASYNC_TO_LDS_B128, ASYNCcnt).
// ------------------------------------------------------------------
__global__ void k_gemm_wmma(const float* __restrict__ X, const float* __restrict__ W,
                            float* __restrict__ H, int nrows) {
  __shared__ float Atile[16][NF + 4];     // +4 pad: kill 16-way bank conflicts on column reads

  const int row0 = blockIdx.x * 16;
  const int tid = threadIdx.x;

  if (row0 + 16 <= nrows) {
    // full tile: async DMA global -> LDS. 8 KB = 512 x 16B chunks, 2 per thread.
    // rows are contiguous in global (stride NF), padded in LDS (stride NF+4).
    const char* gbase = (const char*)(X + (size_t)row0 * NF);
    #pragma unroll
    for (int i = 0; i < 2; ++i) {
      int c = tid + i * 256;                       // chunk id 0..511
      int r = c >> 5;                              // row 0..15 (32 chunks per row)
      int w = c & 31;                              // 16B chunk within row
      unsigned lds  = (unsigned)(uintptr_t)&Atile[r][w * 4];
      unsigned voff = (unsigned)(c * 16);          // contiguous global byte offset
      asm volatile("global_load_async_to_lds_b128 %0, %1, %2"
                   :: "v"(lds), "v"(voff), "s"(gbase) : "memory");
    }
    asm volatile("s_wait_asynccnt 0x0" ::: "memory");
  } else {
    // generic tail fallback (not taken for N % 16 == 0)
    for (int i = 0; i < 8; ++i) {
      int idx = tid * 8 + i;
      int r = idx >> 7;
      int c = idx & (NF - 1);
      int row = row0 + r;
      Atile[r][c] = (row < nrows) ? X[(size_t)row * NF + c] : 0.0f;
    }
  }
  __syncthreads();

  const int wave = tid >> 5;              // 0..7 -> output column tile
  const int lane = tid & 31;
  const int half = lane >> 4;             // 0: lanes 0-15, 1: lanes 16-31
  const int m    = lane & 15;             // row (A/C) or col (B/C) index inside tile
  const int col0 = wave * 16;

  v8f acc = {};
  #pragma unroll
  for (int k0 = 0; k0 < NF; k0 += 4) {
    // A 16x4 fragment (ISA 7.12.2): VGPR0 = K = 2*half, VGPR1 = K = 2*half+1
    v2f a, b;
    a.x = Atile[m][k0 + 2 * half + 0];
    a.y = Atile[m][k0 + 2 * half + 1];
    // B 4x16 fragment: row K striped across lanes within a VGPR (mirror of C layout)
    b.x = W[(size_t)(k0 + 2 * half + 0) * NF + col0 + m];
    b.y = W[(size_t)(k0 + 2 * half + 1) * NF + col0 + m];
    acc = __builtin_amdgcn_wmma_f32_16x16x4_f32(
        /*neg_a=*/false, a, /*neg_b=*/false, b,
        /*c_mod=*/(short)0, acc, /*reuse_a=*/false, /*reuse_b=*/false);
  }

  // C/D 16x16 f32 layout: VGPR v -> M = v + 8*half, N = lane&15
  #pragma unroll
  for (int v = 0; v < 8; ++v) {
    int row = row0 + v + 8 * half;
    if (row < nrows) H[(size_t)row * NF + col0 + m] = acc[v];
  }
}

// ------------------------------------------------------------------
// Out[i,:] = bias + H[i,:] * dis[i]^2   (self-loop contribution + bias)
// one thread per float4 chunk
// ------------------------------------------------------------------
__global__ void k_init_out(const float* __restrict__ H, const float* __restrict__ dis,
                           const float* __restrict__ bias, float* __restrict__ Out, int n) {
  int t = blockIdx.x * blockDim.x + threadIdx.x;
  if (t >= n * 32) return;
  int node = t >> 5;
  int c4 = (t & 31) * 4;
  float dd = dis[node];
  dd *= dd;
  float4 h = *(const float4*)(H + (size_t)node * NF + c4);
  float4 b = *(const float4*)(bias + c4);
  float4 o;
  o.x = b.x + h.x * dd;
  o.y = b.y + h.y * dd;
  o.z = b.z + h.z * dd;
  o.w = b.w + h.w * dd;
  *(float4*)(Out + (size_t)node * NF + c4) = o;
}

// ------------------------------------------------------------------
// edge scatter: Out[dst,:] += H[src,:] * dis[src]*dis[dst]
// one wave32 per edge: lane handles a float4 chunk of the 128-wide row
// ------------------------------------------------------------------
__global__ void k_edge_scatter(const float* __restrict__ H, const int* __restrict__ src,
                               const int* __restrict__ dst, const float* __restrict__ dis,
                               float* __restrict__ Out, int E) {
  int e = blockIdx.x * (blockDim.x >> 5) + (threadIdx.x >> 5);
  int lane = threadIdx.x & 31;
  if (e >= E) return;
  int s = src[e];
  int d = dst[e];
  float w = dis[s] * dis[d];
  float4 v = ((const float4*)(H + (size_t)s * NF))[lane];
  float* o = Out + (size_t)d * NF + lane * 4;
  __builtin_prefetch(o, 0, 1);   // gfx1250 global_prefetch_b8: pull dst row toward L2
  atomicAdd(o + 0, v.x * w);
  atomicAdd(o + 1, v.y * w);
  atomicAdd(o + 2, v.z * w);
  atomicAdd(o + 3, v.w * w);
}

// ------------------------------------------------------------------
// row-wise L2 normalize + ReLU, in place. one wave32 per node.
// ------------------------------------------------------------------
__global__ void k_l2norm_relu(float* __restrict__ H, int n) {
  int node = blockIdx.x * (blockDim.x >> 5) + (threadIdx.x >> 5);
  int lane = threadIdx.x & 31;
  if (node >= n) return;
  float4 v = ((const float4*)(H + (size_t)node * NF))[lane];
  float ss = v.x * v.x + v.y * v.y + v.z * v.z + v.w * v.w;
  #pragma unroll
  for (int off = 16; off > 0; off >>= 1) ss += __shfl_xor(ss, off, 32);
  float s = 1.0f / fmaxf(sqrtf(ss), EPS);
  v.x = fmaxf(v.x * s, 0.0f);
  v.y = fmaxf(v.y * s, 0.0f);
  v.z = fmaxf(v.z * s, 0.0f);
  v.w = fmaxf(v.w * s, 0.0f);
  ((float4*)(H + (size_t)node * NF))[lane] = v;
}

// ------------------------------------------------------------------
// pooling: sums[batch[i],:] += H[i,:]; cnt[batch[i]] += 1
// ------------------------------------------------------------------
__global__ void k_pool(const float* __restrict__ H, const int* __restrict__ batch,
                       float* __restrict__ sums, float* __restrict__ cnt, int n) {
  int node = blockIdx.x * (blockDim.x >> 5) + (threadIdx.x >> 5);
  int lane = threadIdx.x & 31;
  if (node >= n) return;
  int g = batch[node];
  float4 v = ((const float4*)(H + (size_t)node * NF))[lane];
  float* s = sums + (size_t)g * NF + lane * 4;
  atomicAdd(s + 0, v.x);
  atomicAdd(s + 1, v.y);
  atomicAdd(s + 2, v.z);
  atomicAdd(s + 3, v.w);
  if (lane == 0) atomicAdd(&cnt[g], 1.0f);
}

// out[g,c] = bl[c] + sum_k (sums[g,k]/max(cnt[g],1)) * Wl[k,c]   (64x10, tiny)
__global__ void k_final(const float* __restrict__ sums, const float* __restrict__ cnt,
                        const float* __restrict__ Wl, const float* __restrict__ bl,
                        float* __restrict__ out, int ngraph, int nclass) {
  int t = blockIdx.x * blockDim.x + threadIdx.x;
  if (t >= ngraph * nclass) return;
  int g = t / nclass;
  int c = t % nclass;
  float inv = 1.0f / fmaxf(cnt[g], 1.0f);
  float acc = bl[c];
  for (int k = 0; k < NF; ++k)
    acc += sums[(size_t)g * NF + k] * inv * Wl[k * nclass + c];
  out[t] = acc;
}

// ------------------------------------------------------------------
extern "C" void kernel_launch(void* const* d_in, const int* in_sizes, int n_in,
                              void* d_out, int out_size, void* d_ws, size_t ws_size,
                              hipStream_t stream) {
  const float* x   = (const float*)d_in[0];
  const float* W1  = (const float*)d_in[1];
  const float* b1  = (const float*)d_in[2];
  const float* W2  = (const float*)d_in[3];
  const float* b2  = (const float*)d_in[4];
  const float* W3  = (const float*)d_in[5];
  const float* b3  = (const float*)d_in[6];
  const float* Wl  = (const float*)d_in[7];
  const float* bl  = (const float*)d_in[8];
  const int*   ei  = (const int*)d_in[9];
  const int*   bat = (const int*)d_in[10];

  const int N = in_sizes[0] / NF;       // 100000
  const int E = in_sizes[9] / 2;        // 1600000
  const int NG = 64, NC = 10;
  const int* srcp = ei;
  const int* dstp = ei + E;

  // ---- workspace carve-up (all float, 256B aligned) ----
  char* ws = (char*)d_ws;
  size_t off = 0;
  auto take = [&](size_t nfloats) {
    float* p = (float*)(ws + off);
    off += ((nfloats * 4 + 255) / 256) * 256;
    return p;
  };
  float* dis  = take((size_t)N);             // degree then deg^-1/2
  float* sums = take((size_t)NG * NF + NG);  // pooled sums + counts (contiguous)
  float* cnt  = sums + (size_t)NG * NF;
  float* bufH = take((size_t)N * NF);        // GEMM output
  float* bufO = take((size_t)N * NF);        // aggregated output / next layer input

  const int TB = 256;
  const int wavesPerBlock = TB / 32;

  // ---- degree + symmetric norm ----
  k_fill_f32<<<(N + TB - 1) / TB, TB, 0, stream>>>(dis, 1.0f, N);     // self loop
  k_deg_count<<<(E + TB - 1) / TB, TB, 0, stream>>>(dstp, dis, E);
  k_rsqrt_inplace<<<(N + TB - 1) / TB, TB, 0, stream>>>(dis, N);

  // ---- 3 GCN layers ----
  const float* Ws[3] = {W1, W2, W3};
  const float* bs[3] = {b1, b2, b3};
  const float* in = x;
  for (int l = 0; l < 3; ++l) {
    k_gemm_wmma<<<(N + 15) / 16, TB, 0, stream>>>(in, Ws[l], bufH, N);
    k_init_out<<<((size_t)N * 32 + TB - 1) / TB, TB, 0, stream>>>(bufH, dis, bs[l], bufO, N);
    k_edge_scatter<<<(E + wavesPerBlock - 1) / wavesPerBlock, TB, 0, stream>>>(
        bufH, srcp, dstp, dis, bufO, E);
    k_l2norm_relu<<<(N + wavesPerBlock - 1) / wavesPerBlock, TB, 0, stream>>>(bufO, N);
    in = bufO;
  }

  // ---- global mean pool + classifier ----
  k_fill_f32<<<(NG * NF + NG + TB - 1) / TB, TB, 0, stream>>>(sums, 0.0f, NG * NF + NG);
  k_pool<<<(N + wavesPerBlock - 1) / wavesPerBlock, TB, 0, stream>>>(bufO, bat, sums, cnt, N);
  k_final<<<(NG * NC + TB - 1) / TB, TB, 0, stream>>>(sums, cnt, Wl, bl, (float*)d_out, NG, NC);
}